// GPLVM_Bayesian_33483565040052
// MI455X (gfx1250) — compile-verified
//
#include <hip/hip_runtime.h>
#include <hip/hip_bf16.h>
#include <math.h>

typedef __attribute__((ext_vector_type(2))) float v2f;
typedef __attribute__((ext_vector_type(8))) float v8f;

#define NN 2048
#define DD 256
#define QQ 16
#define MMI 64        // inducing points m
#define MM2 4096      // m*m
#define KF 36         // padded feature rank for psi2 GEMM (34 -> 36)

// scalar slots in workspace
#define S_NOISE  0
#define S_BETA   1
#define S_VAR    2
#define S_LOGVAR 3
#define S_KL     4
#define S_YY     5
#define S_AB     6
#define S_LDK    7
#define S_LDC    8

__device__ __forceinline__ float softplus_f(float x) {
    return (x > 20.f) ? x : log1pf(__expf(x));
}

// ---------------- init: zero scalar slots ----------------
__global__ void k_init(float* sc) {
    int i = blockIdx.x * blockDim.x + threadIdx.x;
    if (i < 32) sc[i] = 0.f;
}

// ---------------- scalars: noise_var, beta, var, logvar ----------------
__global__ void k_scalars(const float* noise_raw, const float* variance, float* sc) {
    float nv = softplus_f(noise_raw[0]);
    sc[S_NOISE]  = nv;
    sc[S_BETA]   = 1.f / nv;
    float v = variance[0];
    sc[S_VAR]    = v;
    sc[S_LOGVAR] = logf(v);
}

// ---------------- KL sum and sum(y*y) ----------------
__global__ void k_reduce(const float* y, const float* q_mu, const float* qls, float* sc) {
    __shared__ float sh[2];
    float kl = 0.f, yy = 0.f;
    for (int i = blockIdx.x * blockDim.x + threadIdx.x; i < NN * DD;
         i += gridDim.x * blockDim.x) {
        float yv = y[i];
        yy += yv * yv;
        if (i < NN * QQ) {
            float s  = softplus_f(qls[i]);
            float mu = q_mu[i];
            kl += -logf(s) + 0.5f * (s * s + mu * mu - 1.f);
        }
    }
    for (int o = 16; o > 0; o >>= 1) {
        kl += __shfl_down(kl, o, 32);
        yy += __shfl_down(yy, o, 32);
    }
    if (threadIdx.x == 0) { sh[0] = 0.f; sh[1] = 0.f; }
    __syncthreads();
    if ((threadIdx.x & 31) == 0) { atomicAdd(&sh[0], kl); atomicAdd(&sh[1], yy); }
    __syncthreads();
    if (threadIdx.x == 0) { atomicAdd(&sc[S_KL], sh[0]); atomicAdd(&sc[S_YY], sh[1]); }
}

// ---------------- per-n prep: psi2 A-features, psi1 weights, sum2 ----------------
__global__ void k_prep_n(const float* q_mu, const float* qls, const float* alpha,
                         float* Afeat, float* sum2, float* w1) {
    int n = blockIdx.x * blockDim.x + threadIdx.x;
    if (n >= NN) return;
    float s2acc = 0.f, cwm = 0.f, s3 = 0.f;
    for (int q = 0; q < QQ; q++) {
        float a   = alpha[q];
        float sig = softplus_f(qls[n * QQ + q]);
        float mu  = q_mu[n * QQ + q];
        float d1  = sig * a + 1.f;
        s2acc += logf(d1);
        w1[n * QQ + q] = a / d1;
        float d2 = 2.f * a * sig + 1.f;
        float w  = a / d2;
        Afeat[n * KF + q]      = 2.f * w * mu;   // pairs with zbar
        Afeat[n * KF + 16 + q] = -w;             // pairs with zbar^2
        cwm += w * mu * mu;
        s3  += 0.5f * logf(d2);
    }
    Afeat[n * KF + 32] = 1.f;           // pairs with (4logvar - s1[ij])
    Afeat[n * KF + 33] = -(cwm + s3);   // pairs with 1
    Afeat[n * KF + 34] = 0.f;
    Afeat[n * KF + 35] = 0.f;
    sum2[n] = s2acc;
}

// ---------------- k_mm, s1, psi2 B-features (K-major 36 x 4096) ----------------
__global__ void k_kmm(const float* z, const float* alpha, const float* sc,
                      float* kmm, float* BfeatT) {
    int idx = blockIdx.x * blockDim.x + threadIdx.x;
    if (idx >= MM2) return;
    int i = idx >> 6, j = idx & 63;
    float var = sc[S_VAR], logvar = sc[S_LOGVAR];
    float s = 0.f;
    for (int q = 0; q < QQ; q++) {
        float zi = z[i * QQ + q], zj = z[j * QQ + q];
        float d  = zi - zj;
        s += alpha[q] * d * d;
        float zb = 0.5f * (zi + zj);
        BfeatT[q * MM2 + idx]        = zb;
        BfeatT[(16 + q) * MM2 + idx] = zb * zb;
    }
    kmm[idx] = var * __expf(-0.5f * s);
    float s1 = 0.25f * s;
    BfeatT[32 * MM2 + idx] = 4.f * logvar - s1;
    BfeatT[33 * MM2 + idx] = 1.f;
    BfeatT[34 * MM2 + idx] = 0.f;
    BfeatT[35 * MM2 + idx] = 0.f;
}

// ---------------- psi1 (n x m) ----------------
__global__ void k_psi1(const float* q_mu, const float* z, const float* w1,
                       const float* sum2, const float* sc, float* psi1) {
    int idx = blockIdx.x * blockDim.x + threadIdx.x;
    if (idx >= NN * MMI) return;
    int n = idx >> 6, mi = idx & 63;
    float s = 0.f;
    for (int q = 0; q < QQ; q++) {
        float d = q_mu[n * QQ + q] - z[mi * QQ + q];
        s += w1[n * QQ + q] * d * d;
    }
    psi1[idx] = __expf(2.f * sc[S_LOGVAR] - 0.5f * s - 0.5f * sum2[n]);
}

// ---------------- psi2 via fp32 WMMA: (2048x36)*(36x4096), exp, reduce over n ----
// block owns one 16-column ij-tile; B-fragments are loop-invariant (18 VGPRs).
// 8 waves stride over the 128 n-tiles, accumulate exp-sums in registers,
// cross-wave reduce through LDS; single store per column, no global atomics.
__global__ void k_psi2(const float* Afeat, const float* BfeatT, float* psi2) {
    __shared__ float sh[8][16];
    int lane   = threadIdx.x & 31;
    int wave   = threadIdx.x >> 5;
    int ijbase = blockIdx.x * 16;
    int mrow   = lane & 15;
    int khalf  = (lane >> 4) * 2;   // ISA A/B 32-bit layout: lanes 0-15 K0/K1, 16-31 K2/K3

    // hoist B fragments: constant for this block's columns
    v2f b[9];
#pragma unroll
    for (int s = 0; s < 9; s++) {
        int k = 4 * s + khalf;
        b[s].x = BfeatT[k * MM2 + ijbase + mrow];
        b[s].y = BfeatT[(k + 1) * MM2 + ijbase + mrow];
    }

    float acc = 0.f;
    for (int nt = wave; nt < 128; nt += 8) {
        int nbase = nt * 16;
        // prefetch next n-tile's A rows (gfx1250 global_prefetch_b8)
        if (nt + 8 < 128)
            __builtin_prefetch(&Afeat[((nt + 8) * 16 + mrow) * KF], 0, 1);
        v8f c = {};
#pragma unroll
        for (int s = 0; s < 9; s++) {
            const float* ap = &Afeat[(nbase + mrow) * KF + 4 * s + khalf];
            v2f a;
            a.x = ap[0];
            a.y = ap[1];
            c = __builtin_amdgcn_wmma_f32_16x16x4_f32(false, a, false, b[s],
                                                      (short)0, c, false, false);
        }
#pragma unroll
        for (int r = 0; r < 8; r++) acc += __expf(c[r]);
    }
    // column col = lane&15: rows 0-7 live in lane col, rows 8-15 in lane col+16
    acc += __shfl_xor(acc, 16, 32);
    if (lane < 16) sh[wave][lane] = acc;
    __syncthreads();
    if (threadIdx.x < 16) {
        float tot = 0.f;
#pragma unroll
        for (int w = 0; w < 8; w++) tot += sh[w][threadIdx.x];
        psi2[ijbase + threadIdx.x] = tot;
    }
}

// ---------------- A = psi1^T (64x2048) @ y (2048x256) via fp32 WMMA ----------------
__global__ void k_gemmA(const float* psi1, const float* y, float* Amat) {
    int lane = threadIdx.x & 31;
    int wave = blockIdx.x * (blockDim.x >> 5) + (threadIdx.x >> 5);  // 0..63
    int mi = wave >> 4, ni = wave & 15;
    int mrow  = lane & 15;
    int khalf = (lane >> 4) * 2;
    v8f c = {};
    for (int k0 = 0; k0 < NN; k0 += 4) {
        int k = k0 + khalf;
        // prefetch one 32-K chunk ahead (global_prefetch_b8)
        if ((k0 & 31) == 0 && k0 + 32 < NN) {
            __builtin_prefetch(&psi1[(k + 32) * MMI + mi * 16 + mrow], 0, 1);
            __builtin_prefetch(&y[(k + 32) * DD + ni * 16 + mrow], 0, 1);
        }
        v2f a, b;
        a.x = psi1[k * MMI + mi * 16 + mrow];        // A[M][K] = psi1[K][M]
        a.y = psi1[(k + 1) * MMI + mi * 16 + mrow];
        b.x = y[k * DD + ni * 16 + mrow];            // B[K][N]
        b.y = y[(k + 1) * DD + ni * 16 + mrow];
        c = __builtin_amdgcn_wmma_f32_16x16x4_f32(false, a, false, b,
                                                  (short)0, c, false, false);
    }
    int col   = ni * 16 + mrow;
    int rbase = mi * 16 + (lane >> 4) * 8;   // C layout: lanes>=16 hold rows 8..15
#pragma unroll
    for (int r = 0; r < 8; r++) Amat[(rbase + r) * DD + col] = c[r];
}

// ---------------- cov1 = beta*psi2 + k_mm ----------------
__global__ void k_cov1(const float* psi2, const float* kmm, const float* sc, float* cov1) {
    int i = blockIdx.x * blockDim.x + threadIdx.x;
    if (i < MM2) cov1[i] = sc[S_BETA] * psi2[i] + kmm[i];
}

// ---------------- 64x64 Cholesky + logdet (single block, 64 threads) ----------------
__global__ void k_chol(const float* src, float* L, float* sc, int slot) {
    int tid = threadIdx.x;  // 0..63, owns row tid
    for (int j = 0; j < MMI; j++) L[tid * MMI + j] = src[tid * MMI + j];
    __syncthreads();
    for (int k = 0; k < MMI; k++) {
        if (tid == k) L[k * MMI + k] = sqrtf(L[k * MMI + k]);
        __syncthreads();
        if (tid > k) L[tid * MMI + k] /= L[k * MMI + k];
        __syncthreads();
        if (tid > k) {
            float lik = L[tid * MMI + k];
            for (int j = k + 1; j <= tid; j++)
                L[tid * MMI + j] -= lik * L[j * MMI + k];
        }
        __syncthreads();
    }
    if (tid == 0) {
        float ld = 0.f;
        for (int i = 0; i < MMI; i++) ld += logf(L[i * MMI + i]);
        sc[slot] = 2.f * ld;
    }
}

// ---------------- triangular solve L L^T x = rhs, one thread per column -----------
__global__ void k_solve(const float* L, const float* rhs, float* out, int ncols) {
    int c = blockIdx.x * blockDim.x + threadIdx.x;
    if (c >= ncols) return;
    float x[MMI];
    for (int i = 0; i < MMI; i++) x[i] = rhs[i * ncols + c];
    for (int i = 0; i < MMI; i++) {            // forward: L z = rhs
        float s = x[i];
        for (int j = 0; j < i; j++) s -= L[i * MMI + j] * x[j];
        x[i] = s / L[i * MMI + i];
    }
    for (int i = MMI - 1; i >= 0; i--) {       // backward: L^T x = z
        float s = x[i];
        for (int j = i + 1; j < MMI; j++) s -= L[j * MMI + i] * x[j];
        x[i] = s / L[i * MMI + i];
    }
    for (int i = 0; i < MMI; i++) out[i * ncols + c] = x[i];
}

// ---------------- sum(A .* B) ----------------
__global__ void k_sumAB(const float* A, const float* B, float* sc) {
    __shared__ float sh;
    float acc = 0.f;
    for (int i = blockIdx.x * blockDim.x + threadIdx.x; i < MMI * DD;
         i += gridDim.x * blockDim.x)
        acc += A[i] * B[i];
    for (int o = 16; o > 0; o >>= 1) acc += __shfl_down(acc, o, 32);
    if (threadIdx.x == 0) sh = 0.f;
    __syncthreads();
    if ((threadIdx.x & 31) == 0) atomicAdd(&sh, acc);
    __syncthreads();
    if (threadIdx.x == 0) atomicAdd(&sc[S_AB], sh);
}

// ---------------- final ELBO assembly ----------------
__global__ void k_final(const float* sc, const float* Xs, float* out) {
    float beta = sc[S_BETA], var = sc[S_VAR];
    float kl_term = sc[S_KL] / (float)(NN * DD);
    float tr_yWy  = beta * sc[S_YY] - sc[S_AB];
    float tr = 0.f;
    for (int i = 0; i < MMI; i++) tr += Xs[i * MMI + i];
    float psi0 = (float)NN * var;
    float F = 0.5f * (float)NN * logf(beta)
            + 0.5f * sc[S_LDK]
            - 0.5f * (float)NN * logf(3.14159265358979323846f)
            - 0.5f * sc[S_LDC]
            - 0.5f * beta * psi0
            + 0.5f * tr;
    F = (F * (float)DD - 0.5f * tr_yWy) / (float)(NN * DD);
    out[0] = F - kl_term;
}

extern "C" void kernel_launch(void* const* d_in, const int* in_sizes, int n_in,
                              void* d_out, int out_size, void* d_ws, size_t ws_size,
                              hipStream_t stream) {
    const float* y         = (const float*)d_in[0];
    const float* q_mu      = (const float*)d_in[1];
    const float* qls       = (const float*)d_in[2];
    const float* z         = (const float*)d_in[3];
    const float* noise_raw = (const float*)d_in[4];
    const float* alpha     = (const float*)d_in[5];
    const float* variance  = (const float*)d_in[6];
    float* out = (float*)d_out;

    float* ws     = (float*)d_ws;
    float* sc     = ws;                   // 32 scalars
    float* kmm    = sc + 32;              // 4096
    float* cholK  = kmm + MM2;            // 4096
    float* cov1   = cholK + MM2;          // 4096
    float* cholC  = cov1 + MM2;           // 4096
    float* psi2   = cholC + MM2;          // 4096
    float* Xsolve = psi2 + MM2;           // 4096  (k_mm^{-1} psi2)
    float* psi1   = Xsolve + MM2;         // N*M = 131072
    float* Amat   = psi1 + NN * MMI;      // 64*256
    float* Bmat   = Amat + MMI * DD;      // 64*256 (cov1^{-1} A)
    float* Afeat  = Bmat + MMI * DD;      // N*36
    float* BfeatT = Afeat + NN * KF;      // 36*4096
    float* sum2   = BfeatT + KF * MM2;    // N
    float* w1     = sum2 + NN;            // N*Q

    k_init   <<<1, 32, 0, stream>>>(sc);
    k_scalars<<<1, 1, 0, stream>>>(noise_raw, variance, sc);
    k_reduce <<<64, 256, 0, stream>>>(y, q_mu, qls, sc);
    k_prep_n <<<8, 256, 0, stream>>>(q_mu, qls, alpha, Afeat, sum2, w1);
    k_kmm    <<<16, 256, 0, stream>>>(z, alpha, sc, kmm, BfeatT);
    k_psi1   <<<512, 256, 0, stream>>>(q_mu, z, w1, sum2, sc, psi1);
    k_psi2   <<<256, 256, 0, stream>>>(Afeat, BfeatT, psi2);
    k_gemmA  <<<8, 256, 0, stream>>>(psi1, y, Amat);
    k_cov1   <<<16, 256, 0, stream>>>(psi2, kmm, sc, cov1);
    k_chol   <<<1, 64, 0, stream>>>(kmm, cholK, sc, S_LDK);
    k_chol   <<<1, 64, 0, stream>>>(cov1, cholC, sc, S_LDC);
    k_solve  <<<1, 64, 0, stream>>>(cholK, psi2, Xsolve, MMI);
    k_solve  <<<1, 256, 0, stream>>>(cholC, Amat, Bmat, DD);
    k_sumAB  <<<16, 256, 0, stream>>>(Amat, Bmat, sc);
    k_final  <<<1, 1, 0, stream>>>(sc, Xsolve, out);
}